// SemanticLevelContext_47502338293806
// MI455X (gfx1250) — compile-verified
//
#include <hip/hip_runtime.h>
#include <hip/hip_bf16.h>

// ---------------------------------------------------------------------------
// Problem constants (from reference): x [B,C,H,W], preds [B,K,H,W]
// ---------------------------------------------------------------------------
#define BB    4
#define CC    256
#define KCLS  6
#define NPIX  65536          // 256*256
#define TOPK  8
#define RBINS 48             // KCLS*TOPK
#define NBINS 2048           // histogram bins over float bits (cert >= 0 -> monotone)
#define TN    1024           // pixels per workgroup in main pass
#define KC2   32             // pixels per LDS chunk (K-tile), double buffered
#define KP2   36             // padded LDS row stride (conflict-free transposed reads)
#define NCH   (TN / KC2)     // 32 chunks per workgroup

typedef __attribute__((ext_vector_type(2))) float v2f;
typedef __attribute__((ext_vector_type(8))) float v8f;
typedef __attribute__((ext_vector_type(4))) int v4i;
typedef v4i __attribute__((address_space(1)))* gv4i_p;   // global v4i*
typedef v4i __attribute__((address_space(3)))* lv4i_p;   // LDS v4i*

#if __has_builtin(__builtin_amdgcn_global_load_async_to_lds_b128) && \
    __has_builtin(__builtin_amdgcn_s_wait_asynccnt)
#define USE_ASYNC 1
#else
#define USE_ASYNC 0
#endif

__device__ __forceinline__ v8f wmma_f32_16x16x4(v2f a, v2f b, v8f c) {
#if __has_builtin(__builtin_amdgcn_wmma_f32_16x16x4_f32)
  return __builtin_amdgcn_wmma_f32_16x16x4_f32(false, a, false, b, (short)0, c,
                                               false, false);
#else
  c[0] += a.x * b.x; c[1] += a.y * b.y;   // fallback so histogram reveals it
  return c;
#endif
}

// Stage one 32-pixel x 256-channel chunk of x into an LDS buffer.
// Async path: GLOBAL_LOAD_ASYNC_TO_LDS_B128 (ASYNCcnt), no VGPR staging.
__device__ __forceinline__ void issue_chunk(const float* __restrict__ xb,
                                            float* xsbuf, int n0, int tid) {
#pragma unroll
  for (int i = 0; i < 8; ++i) {
    const int c = (tid >> 3) + i * 32;
    const int k4 = (tid & 7) << 2;
    const float* g = xb + (size_t)c * NPIX + n0 + k4;
    float* l = &xsbuf[c * KP2 + k4];
#if USE_ASYNC
    __builtin_amdgcn_global_load_async_to_lds_b128((gv4i_p)(void*)g,
                                                   (lv4i_p)(void*)l, 0, 0);
#else
    const float4 v = *(const float4*)g;
    *(float4*)l = v;
    __builtin_prefetch(g + KC2, 0, 1);
#endif
  }
}

// ---------------------------------------------------------------------------
// Kernel 0: zero the accumulator region of the workspace
// ---------------------------------------------------------------------------
__global__ void k_zero(unsigned* __restrict__ p, int n) {
  int i = blockIdx.x * 256 + threadIdx.x;
  if (i < n) p[i] = 0u;
}

// ---------------------------------------------------------------------------
// Kernel 1: per-pixel certainty (top1-top2), argmax class, per-(b,class)
// histogram of cert float bits (monotone since cert >= 0), atomic max bits.
// grid = B * (NPIX/256), block = 256
// ---------------------------------------------------------------------------
__global__ __launch_bounds__(256) void k_cert(
    const float* __restrict__ preds, float* __restrict__ cert,
    unsigned* __restrict__ argm, unsigned* __restrict__ hist,
    unsigned* __restrict__ maxb) {
  __shared__ unsigned lh[KCLS * NBINS];   // 48 KB
  __shared__ unsigned lmax[KCLS];
  const int tid = threadIdx.x;
  for (int i = tid; i < KCLS * NBINS; i += 256) lh[i] = 0u;
  if (tid < KCLS) lmax[tid] = 0u;
  __syncthreads();

  const int b = blockIdx.x >> 8;
  const int n = ((blockIdx.x & 255) << 8) + tid;
  const float* p = preds + ((size_t)b * KCLS) * NPIX + n;

  float m1 = -3.0e38f, m2 = -3.0e38f;
  int am = 0;
#pragma unroll
  for (int k = 0; k < KCLS; ++k) {
    float v = p[(size_t)k * NPIX];
    if (v > m1) { m2 = m1; m1 = v; am = k; }
    else if (v > m2) { m2 = v; }
  }
  const float ce = m1 - m2;                 // >= 0
  const unsigned bits = __float_as_uint(ce);
  const int bin = (bits >> 20) & (NBINS - 1);
  atomicAdd(&lh[am * NBINS + bin], 1u);
  atomicMax(&lmax[am], bits);
  cert[(size_t)b * NPIX + n] = ce;
  argm[(size_t)b * NPIX + n] = (unsigned)am;
  __syncthreads();

  for (int i = tid; i < KCLS * NBINS; i += 256) {
    unsigned v = lh[i];
    if (v) atomicAdd(&hist[(size_t)b * KCLS * NBINS + i], v);
  }
  if (tid < KCLS) atomicMax(&maxb[b * KCLS + tid], lmax[tid]);
}

// ---------------------------------------------------------------------------
// Kernel 2: rank-select thresholds per (b, class) from histogram.
// ---------------------------------------------------------------------------
__global__ void k_thr(const unsigned* __restrict__ hist,
                      const unsigned* __restrict__ maxb,
                      float* __restrict__ thr, int* __restrict__ mcnt,
                      float* __restrict__ maxc) {
  const int t = threadIdx.x;
  if (t >= BB * KCLS) return;
  const unsigned* h = hist + (size_t)t * NBINS;

  long m = 0;
  for (int i = 0; i < NBINS; ++i) m += (long)h[i];
  mcnt[t] = (int)m;
  maxc[t] = __uint_as_float(maxb[t]);

  long rank[TOPK];
#pragma unroll
  for (int tt = 1; tt <= TOPK; ++tt) {
    long ks = (m * tt) >> 3;                // (m*t)/topk
    rank[tt - 1] = (ks > 0) ? (ks - 1) : 0;
  }
  int ti = 0;
  long cum = 0;
  for (int bin = NBINS - 1; bin >= 0 && ti < TOPK; --bin) {
    cum += (long)h[bin];
    while (ti < TOPK && cum > rank[ti]) {
      thr[t * TOPK + ti] = __uint_as_float((unsigned)bin << 20);
      ++ti;
    }
  }
  while (ti < TOPK) { thr[t * TOPK + ti] = __uint_as_float(0x7F000000u); ++ti; }
}

// ---------------------------------------------------------------------------
// Kernel 3: main pass. One WG = 1024 pixels x all 256 channels of one batch.
// Prepass computes per-pixel (bin r, weight e) for the tile. Then a double-
// buffered pipeline: async-load chunk n+1 to LDS while WMMAing chunk n.
//   A (16x4) = one-hot bin/weight matrix, B (4x16) = X tile (transposed read)
//   D += A*B via V_WMMA_F32_16X16X4_F32, 6 resident accumulators per wave.
// grid = B * (NPIX/TN) = 256, block = 256 (8 waves)
// ---------------------------------------------------------------------------
__global__ __launch_bounds__(256) void k_main(
    const float* __restrict__ x, const float* __restrict__ cert,
    const unsigned* __restrict__ argm, const float* __restrict__ thr,
    const float* __restrict__ maxc, float* __restrict__ P,
    float* __restrict__ Z) {
  __shared__ __align__(16) float Xs[2][CC * KP2];  // 2 * 36864 B
  __shared__ float esAll[TN];
  __shared__ int rsAll[TN];
  __shared__ float zloc[RBINS];
  __shared__ float thrS[KCLS][TOPK];
  __shared__ float maxcS[KCLS];

  const int tid = threadIdx.x;
  const int b = blockIdx.x >> 6;           // 64 tiles per batch
  const int nbase = (blockIdx.x & 63) * TN;
  const float* xb = x + (size_t)b * CC * NPIX;

  // kick off DMA of chunk 0 immediately; it needs nothing from LDS
  issue_chunk(xb, &Xs[0][0], nbase, tid);

  if (tid < KCLS * TOPK) thrS[tid >> 3][tid & 7] = thr[b * KCLS * TOPK + tid];
  if (tid < KCLS) maxcS[tid] = maxc[b * KCLS + tid];
  if (tid < RBINS) zloc[tid] = 0.0f;
  __syncthreads();

  // prepass: per-pixel bin + softmax weight for the whole tile (overlaps DMA)
#pragma unroll
  for (int i = 0; i < TN / 256; ++i) {
    const int px = i * 256 + tid;
    const int n = nbase + px;
    const float ce = cert[(size_t)b * NPIX + n];
    const int c = (int)argm[(size_t)b * NPIX + n];
    int tm = TOPK - 1;
#pragma unroll
    for (int t = 0; t < TOPK; ++t) {
      if (ce >= thrS[c][t]) { tm = t; break; }   // thresholds decrease in t
    }
    const float e = __expf(ce - maxcS[c]);
    rsAll[px] = c * TOPK + tm;
    esAll[px] = e;
    atomicAdd(&zloc[c * TOPK + tm], e);
  }

  const int wave = tid >> 5;
  const int lane = tid & 31;
  const int half = lane >> 4;   // 0: lanes 0-15, 1: lanes 16-31
  const int lm = lane & 15;
  const int nt0 = wave * 2;     // channel tiles {nt0, nt0+1}

  v8f acc[3][2];
  const v8f z8 = {0.f, 0.f, 0.f, 0.f, 0.f, 0.f, 0.f, 0.f};
#pragma unroll
  for (int i = 0; i < 3; ++i)
#pragma unroll
    for (int j = 0; j < 2; ++j) acc[i][j] = z8;

  for (int ch = 0; ch < NCH; ++ch) {
    const int cur = ch & 1;
    if (ch + 1 < NCH) {
      // start DMA of next chunk into the other buffer, then wait until only
      // those 8 remain outstanding -> this chunk's loads (in-order) retired
      issue_chunk(xb, &Xs[1 - cur][0], nbase + (ch + 1) * KC2, tid);
#if USE_ASYNC
      __builtin_amdgcn_s_wait_asynccnt(8);
#endif
    } else {
#if USE_ASYNC
      __builtin_amdgcn_s_wait_asynccnt(0);
#endif
    }
    __syncthreads();   // everyone's chunk `ch` data visible in LDS

    const float* xsc = &Xs[cur][0];
    const int kbase = ch * KC2;
    for (int ks = 0; ks < KC2; ks += 4) {
      // A operand: lanes 0-15 hold K={ks,ks+1}, lanes 16-31 K={ks+2,ks+3}
      const int kA = kbase + ks + half * 2;
      const int r0 = rsAll[kA];
      const int r1 = rsAll[kA + 1];
      const float e0 = esAll[kA];
      const float e1 = esAll[kA + 1];

      // B operand: VGPR0 = K {ks,ks+1}, VGPR1 = K {ks+2,ks+3}; N col = channel
      v2f bm[2];
#pragma unroll
      for (int j = 0; j < 2; ++j) {
        const int c = (nt0 + j) * 16 + lm;
        bm[j].x = xsc[c * KP2 + ks + half];
        bm[j].y = xsc[c * KP2 + ks + 2 + half];
      }
#pragma unroll
      for (int mt = 0; mt < 3; ++mt) {
        const int row = mt * 16 + lm;
        v2f a;
        a.x = (r0 == row) ? e0 : 0.0f;
        a.y = (r1 == row) ? e1 : 0.0f;
#pragma unroll
        for (int j = 0; j < 2; ++j)
          acc[mt][j] = wmma_f32_16x16x4(a, bm[j], acc[mt][j]);
      }
    }
    __syncthreads();   // done reading buffer before next iteration overwrites
  }

  // epilogue: D layout -> lanes 0-15: VGPR q = M q; lanes 16-31: M = 8+q
#pragma unroll
  for (int mt = 0; mt < 3; ++mt) {
#pragma unroll
    for (int j = 0; j < 2; ++j) {
      const int row0 = mt * 16 + half * 8;
      const int col = (nt0 + j) * 16 + lm;
#pragma unroll
      for (int q = 0; q < 8; ++q) {
        atomicAdd(&P[(size_t)(b * RBINS + row0 + q) * CC + col],
                  acc[mt][j][q]);
      }
    }
  }
  if (tid < RBINS) atomicAdd(&Z[b * RBINS + tid], zloc[tid]);
}

// ---------------------------------------------------------------------------
// Kernel 4: finalize. Nested selection -> cumulative sums over t, normalize.
// ---------------------------------------------------------------------------
__global__ __launch_bounds__(256) void k_final(const float* __restrict__ P,
                                               const float* __restrict__ Z,
                                               const int* __restrict__ mcnt,
                                               float* __restrict__ out) {
  const int bk = blockIdx.x;      // b*KCLS + c
  const int b = bk / KCLS;
  const int c = bk % KCLS;
  const int chn = threadIdx.x;
  const int m = mcnt[bk];

  float cz[TOPK];
  float zacc = 0.0f;
#pragma unroll
  for (int j = 0; j < TOPK; ++j) {
    zacc += Z[b * RBINS + c * TOPK + j];
    cz[j] = zacc;
  }
  float cp[TOPK];
  float pacc = 0.0f;
#pragma unroll
  for (int j = 0; j < TOPK; ++j) {
    pacc += P[(size_t)(b * RBINS + c * TOPK + j) * CC + chn];
    cp[j] = pacc;
  }

  const size_t fsBase = (size_t)(b * CC + chn) * (KCLS * TOPK);
  const size_t fgBase =
      (size_t)BB * CC * KCLS * TOPK + (size_t)(b * CC + chn) * KCLS;
#pragma unroll
  for (int j = 0; j < TOPK; ++j) {
    float v;
    if (m <= 0) v = 0.0f;
    else if (m >= TOPK) v = (cz[j] > 0.0f) ? cp[j] / cz[j] : 0.0f;
    else v = (cz[TOPK - 1] > 0.0f) ? cp[TOPK - 1] / cz[TOPK - 1] : 0.0f;
    const int r = c * TOPK + j;
    out[fsBase + r] = v;
    if (r >= 42) out[fgBase + (r - 42)] = v;   // fg = fs rows 42..47
  }
}

// ---------------------------------------------------------------------------
extern "C" void kernel_launch(void* const* d_in, const int* in_sizes, int n_in,
                              void* d_out, int out_size, void* d_ws,
                              size_t ws_size, hipStream_t stream) {
  const float* x = (const float*)d_in[0];       // [4,256,256,256]
  const float* preds = (const float*)d_in[1];   // [4,6,256,256]
  float* out = (float*)d_out;

  char* ws = (char*)d_ws;
  float* certp = (float*)(ws);                          // 1,048,576 B
  unsigned* argmp = (unsigned*)(ws + 1048576);          // 1,048,576 B
  char* zbase = ws + 2097152;                           // zeroed region:
  unsigned* histp = (unsigned*)(zbase);                 //   196,608 B
  unsigned* maxbp = (unsigned*)(zbase + 196608);        //       256 B
  int* mcntp = (int*)(zbase + 196864);                  //       256 B
  float* thrp = (float*)(zbase + 197120);               //     1,024 B
  float* maxcp = (float*)(zbase + 198144);              //       256 B
  float* Pp = (float*)(zbase + 198400);                 //   196,608 B
  float* Zp = (float*)(zbase + 395008);                 //     1,024 B
  const int zero_words = 396032 / 4;                    // 99,008

  k_zero<<<(zero_words + 255) / 256, 256, 0, stream>>>((unsigned*)zbase,
                                                       zero_words);
  k_cert<<<BB * (NPIX / 256), 256, 0, stream>>>(preds, certp, argmp, histp,
                                                maxbp);
  k_thr<<<1, 32, 0, stream>>>(histp, maxbp, thrp, mcntp, maxcp);
  k_main<<<BB * (NPIX / TN), 256, 0, stream>>>(x, certp, argmp, thrp, maxcp,
                                               Pp, Zp);
  k_final<<<BB * KCLS, 256, 0, stream>>>(Pp, Zp, mcntp, out);
}